// ResidualTemporalConvLayer_3332894621905
// MI455X (gfx1250) — compile-verified
//
#include <hip/hip_runtime.h>

typedef __attribute__((ext_vector_type(2))) float v2f;
typedef __attribute__((ext_vector_type(4))) float v4f;
typedef __attribute__((ext_vector_type(8))) float v8f;

#define B_N   2048
#define C_INN 64
#define C_OUTN 64
#define T_N   32
#define E_N   32768
#define EPSF  1e-5f
#define SLOPEF 0.01f
#define HN    (B_N * T_N * C_OUTN)   // 4194304 floats per (B,T,C) buffer
#define WSTRIDE 66                   // padded LDS stride for transposed weights

// ---------------------------------------------------------------- zero fill
__global__ void k_zero(float* __restrict__ p, long n) {
    long i = (long)blockIdx.x * blockDim.x + threadIdx.x;
    long stride = (long)gridDim.x * blockDim.x;
    for (; i < n; i += stride) p[i] = 0.0f;
}

// ------------------------------------- conv1d + leakyrelu + layernorm + BN stats
// one block (128 threads) per sample b; writes h in (B,T,C) layout
__global__ __launch_bounds__(128) void k_conv_ln(
    const float* __restrict__ x,   const float* __restrict__ cw,
    const float* __restrict__ cb,  const float* __restrict__ lng,
    const float* __restrict__ lnb, float* __restrict__ h,
    float* __restrict__ bnsum, float* __restrict__ bnsq) {
    __shared__ float xs[C_INN * T_N];      // 8 KB input tile
    __shared__ float red[128], red2[128];
    __shared__ float csum[C_OUTN], csq[C_OUTN];
    __shared__ float s_mu, s_rs;

    const int b = blockIdx.x, tid = threadIdx.x;
    for (int i = tid; i < C_INN * T_N; i += 128)
        xs[i] = x[(long)b * (C_INN * T_N) + i];
    if (tid < C_OUTN) { csum[tid] = 0.0f; csq[tid] = 0.0f; }
    __syncthreads();

    float y[16];
    float psum = 0.0f, psq = 0.0f;
#pragma unroll
    for (int j = 0; j < 16; ++j) {
        const int idx = tid + 128 * j;          // idx == o*32 + t
        const int o = idx >> 5, t = idx & 31;
        float acc = cb[o];
        const float* w = cw + o * (C_INN * 3);
        for (int i = 0; i < C_INN; ++i) {
            const float w0 = w[i * 3 + 0], w1 = w[i * 3 + 1], w2 = w[i * 3 + 2];
            float a = xs[i * 32 + t] * w2;
            if (t >= 1) a += xs[i * 32 + t - 1] * w1;
            if (t >= 2) a += xs[i * 32 + t - 2] * w0;
            acc += a;
        }
        acc = (acc >= 0.0f) ? acc : SLOPEF * acc;   // leaky relu
        y[j] = acc; psum += acc; psq += acc * acc;
    }
    red[tid] = psum; red2[tid] = psq;
    __syncthreads();
    for (int s = 64; s > 0; s >>= 1) {
        if (tid < s) { red[tid] += red[tid + s]; red2[tid] += red2[tid + s]; }
        __syncthreads();
    }
    if (tid == 0) {
        const float mu  = red[0] * (1.0f / 2048.0f);
        const float var = red2[0] * (1.0f / 2048.0f) - mu * mu;
        s_mu = mu; s_rs = rsqrtf(var + EPSF);
    }
    __syncthreads();
    const float mu = s_mu, rs = s_rs;
#pragma unroll
    for (int j = 0; j < 16; ++j) {
        const int idx = tid + 128 * j;
        const int o = idx >> 5, t = idx & 31;
        const float hv = (y[j] - mu) * rs * lng[idx] + lnb[idx];
        h[((long)b * T_N + t) * C_OUTN + o] = hv;   // (B,T,C) layout
        atomicAdd(&csum[o], hv);
        atomicAdd(&csq[o], hv * hv);
    }
    __syncthreads();
    if (tid < C_OUTN) {
        atomicAdd(&bnsum[tid], csum[tid]);
        atomicAdd(&bnsq[tid], csq[tid]);
    }
}

// -------------------------- BN finalize -> folded affine (scale, shift)
__global__ void k_bnstats(const float* __restrict__ bnsum,
                          const float* __restrict__ bnsq,
                          const float* __restrict__ bng,
                          const float* __restrict__ bnb,
                          float* __restrict__ bsc, float* __restrict__ bsh) {
    const int c = threadIdx.x;
    if (c < C_OUTN) {
        const float n = (float)(B_N * T_N);
        const float m = bnsum[c] / n;
        const float v = bnsq[c] / n - m * m;
        const float r = rsqrtf(v + EPSF);
        const float sc = r * bng[c];
        bsc[c] = sc;
        bsh[c] = bnb[c] - m * sc;
    }
}

// ------------- hop1: p1[dst] += w * lrelu(h[src]*sc + sh) per (t,c)
__global__ __launch_bounds__(256) void k_hop1(
    const float* __restrict__ h, const long long* __restrict__ ei,
    const float* __restrict__ ew, const float* __restrict__ bsc,
    const float* __restrict__ bsh, float* __restrict__ p1) {
    const int e = blockIdx.x, tid = threadIdx.x;
    const long long src = ei[e], dst = ei[E_N + e];
    const float w = ew[e];
    const int c = tid & 63;                      // constant per thread across j
    const float sc = bsc[c], sh = bsh[c];
    const float* zs = h + src * (long)(T_N * C_OUTN);
    float* pd = p1 + dst * (long)(T_N * C_OUTN);
#pragma unroll
    for (int j = 0; j < 8; ++j) {
        const int idx = tid + 256 * j;           // == t*64 + c
        float z = zs[idx] * sc + sh;
        z = (z >= 0.0f) ? z : SLOPEF * z;
        atomicAdd(&pd[idx], w * z);
    }
}

// ---------------------------- hop2: p2[dst] += w * p1[src]
__global__ __launch_bounds__(256) void k_hop2(
    const float* __restrict__ p1, const long long* __restrict__ ei,
    const float* __restrict__ ew, float* __restrict__ p2) {
    const int e = blockIdx.x, tid = threadIdx.x;
    const long long src = ei[e], dst = ei[E_N + e];
    const float w = ew[e];
    const float* zs = p1 + src * (long)(T_N * C_OUTN);
    float* pd = p2 + dst * (long)(T_N * C_OUTN);
#pragma unroll
    for (int j = 0; j < 8; ++j) {
        const int idx = tid + 256 * j;
        atomicAdd(&pd[idx], w * zs[idx]);
    }
}

// -------- fused TAG GEMM (fp32 WMMA) + bias + residual + (B,C,T) transpose
// out[b,g,t] = h[b,t,g] + tag_b[g] + sum_s  S_s[b,t,:] @ W_s[:,g]
//   S_0 = lrelu(h*sc + sh), S_1 = p1, S_2 = p2
__global__ __launch_bounds__(256) void k_tag_gemm(
    const float* __restrict__ h,  const float* __restrict__ p1,
    const float* __restrict__ p2, const float* __restrict__ tagw,
    const float* __restrict__ tagb, const float* __restrict__ bsc,
    const float* __restrict__ bsh, float* __restrict__ out) {
    // weights transposed in LDS: Ws[s][n*WSTRIDE + f]  (one b64 per B-fragment)
    __shared__ __attribute__((aligned(16))) float Ws[3 * 64 * WSTRIDE];
    __shared__ __attribute__((aligned(16))) float sc_s[C_OUTN];
    __shared__ __attribute__((aligned(16))) float sh_s[C_OUTN];
    const int tid = threadIdx.x;
    for (int i = tid; i < 3 * 64 * 64; i += 256) {
        const int s = i >> 12, rem = i & 4095;
        const int f = rem >> 6, n = rem & 63;
        Ws[s * (64 * WSTRIDE) + n * WSTRIDE + f] = tagw[i];
    }
    if (tid < C_OUTN) { sc_s[tid] = bsc[tid]; sh_s[tid] = bsh[tid]; }
    __syncthreads();

    const int wave = tid >> 5, lane = tid & 31;
    const long rowbase = (long)blockIdx.x * 128 + wave * 16;
    const int mrow = lane & 15;                  // M (A) / N (B/C/D) index
    const int koff = (lane >> 4) << 1;           // lanes 16-31 hold K+2,K+3
    const long arow = rowbase + mrow;

    v8f acc[4] = {};                             // 4 N-tiles of 16x16 f32

#pragma unroll
    for (int s = 0; s < 3; ++s) {
        const float* src = (s == 0) ? h : (s == 1) ? p1 : p2;
        const float* W = Ws + s * (64 * WSTRIDE);
#pragma unroll
        for (int kk = 0; kk < 16; ++kk) {
            const int kc = 4 * kk + koff;
            // A fragment: 16x4 f32 (ISA layout: v0=K, v1=K+1 per lane half)
            v2f a = *(const v2f*)(src + arow * 64 + kc);
            if (s == 0) {                        // folded BN + leaky relu
                const v2f sc = *(const v2f*)(sc_s + kc);
                const v2f sh = *(const v2f*)(sh_s + kc);
                const float z0 = a.x * sc.x + sh.x;
                const float z1 = a.y * sc.y + sh.y;
                a.x = (z0 >= 0.0f) ? z0 : SLOPEF * z0;
                a.y = (z1 >= 0.0f) ? z1 : SLOPEF * z1;
            }
#pragma unroll
            for (int nt = 0; nt < 4; ++nt) {
                // B fragment: contiguous pair (f=kc, kc+1) in transposed LDS
                const v2f bfrag = *(const v2f*)(W + (nt * 16 + mrow) * WSTRIDE + kc);
                acc[nt] = __builtin_amdgcn_wmma_f32_16x16x4_f32(
                    false, a, false, bfrag, (short)0, acc[nt], false, false);
            }
        }
    }

    // epilogue: D layout VGPR j -> M=j (lanes 0-15) / M=j+8 (lanes 16-31)
    const int off8 = (lane >> 4) << 3;
#pragma unroll
    for (int nt = 0; nt < 4; ++nt) {
        const int g = nt * 16 + mrow;
        const float tb = tagb[g];
#pragma unroll
        for (int jh = 0; jh < 2; ++jh) {
            const long r0 = rowbase + off8 + jh * 4;   // 4 consecutive rows
            v4f val;
#pragma unroll
            for (int j = 0; j < 4; ++j)
                val[j] = acc[nt][jh * 4 + j] + tb + h[(r0 + j) * 64 + g];
            const long bb = r0 >> 5;                   // sample index
            const int t = (int)(r0 & 31);              // 4-aligned -> 16B store
            *(v4f*)(out + (bb * 64 + g) * 32 + t) = val;
        }
    }
}

extern "C" void kernel_launch(void* const* d_in, const int* in_sizes, int n_in,
                              void* d_out, int out_size, void* d_ws, size_t ws_size,
                              hipStream_t stream) {
    const float*      x    = (const float*)d_in[0];
    const long long*  ei   = (const long long*)d_in[1];
    const float*      ew   = (const float*)d_in[2];
    const float*      cw   = (const float*)d_in[3];
    const float*      cb   = (const float*)d_in[4];
    const float*      lng  = (const float*)d_in[5];
    const float*      lnb  = (const float*)d_in[6];
    const float*      bng  = (const float*)d_in[7];
    const float*      bnb  = (const float*)d_in[8];
    const float*      tagw = (const float*)d_in[9];
    const float*      tagb = (const float*)d_in[10];

    float* ws = (float*)d_ws;
    float* h     = ws;                 // HN floats  (B,T,C)
    float* p1    = ws + (long)HN;      // HN floats
    float* p2    = ws + 2L * HN;       // HN floats
    float* stats = ws + 3L * HN;       // 256 floats
    float* bnsum = stats;
    float* bnsq  = stats + 64;
    float* bsc   = stats + 128;        // folded BN scale
    float* bsh   = stats + 192;        // folded BN shift

    // zero p1, p2, bnsum, bnsq (contiguous region)
    k_zero<<<2048, 256, 0, stream>>>(p1, 2L * HN + 128);
    k_conv_ln<<<B_N, 128, 0, stream>>>(x, cw, cb, lng, lnb, h, bnsum, bnsq);
    k_bnstats<<<1, 64, 0, stream>>>(bnsum, bnsq, bng, bnb, bsc, bsh);
    k_hop1<<<E_N, 256, 0, stream>>>(h, ei, ew, bsc, bsh, p1);
    k_hop2<<<E_N, 256, 0, stream>>>(p1, ei, ew, p2);
    k_tag_gemm<<<(B_N * T_N) / 128, 256, 0, stream>>>(
        h, p1, p2, tagw, tagb, bsc, bsh, (float*)d_out);
}